// TopExpertsRouter_5918464934128
// MI455X (gfx1250) — compile-verified
//
#include <hip/hip_runtime.h>
#include <math.h>
#include <stdint.h>

typedef __attribute__((ext_vector_type(2))) float v2f;
typedef __attribute__((ext_vector_type(8))) float v8f;

#define D_MODEL   4096
#define N_EXPERTS 64
#define TOPK      8
#define M_TILE    64     // tokens per workgroup
#define KC        64     // K-chunk staged in LDS per stage
#define NSTAGES   (D_MODEL / KC)   // 64
#define LSTRIDE   68     // padded row stride (floats): 272B rows -> 16B aligned, bank-advance 4
#define ESTRIDE   65     // logits row stride in epilogue
#define BLOCK     128    // 4 wave32
#define BUF_FLOATS ((M_TILE + N_EXPERTS) * LSTRIDE)   // 8704 floats per stage buffer
#define EPS       1e-9f

// Async DMA: one 16-byte global->LDS copy per lane, tracked by ASYNCcnt.
__device__ __forceinline__ void async_b128(const float* gptr, float* lptr) {
    uint32_t lds = (uint32_t)(uintptr_t)lptr;   // generic low 32 bits == LDS offset
    asm volatile("global_load_async_to_lds_b128 %0, %1, off"
                 :: "v"(lds), "v"(gptr)
                 : "memory");
}

__device__ __forceinline__ void issue_stage(const float* __restrict__ x,
                                            const float* __restrict__ W,
                                            int tok_base, int n_tokens, int kb,
                                            float* Xs, float* Ws, int tid)
{
    // X tile: 64 rows x 64 cols, 16B per lane, 8 instructions/thread, coalesced.
    #pragma unroll
    for (int i = 0; i < (M_TILE * (KC / 4)) / BLOCK; ++i) {   // 8
        int idx = tid + i * BLOCK;
        int r   = idx >> 4;            // 16 float4 per row
        int c4  = idx & 15;
        int row = tok_base + r;
        if (row >= n_tokens) row = n_tokens - 1;
        async_b128(x + (size_t)row * D_MODEL + kb + c4 * 4,
                   Xs + r * LSTRIDE + c4 * 4);
    }
    // W tile: 64 experts x 64 cols (L2-resident), 8 instructions/thread.
    #pragma unroll
    for (int i = 0; i < (N_EXPERTS * (KC / 4)) / BLOCK; ++i) { // 8
        int idx = tid + i * BLOCK;
        int e   = idx >> 4;
        int c4  = idx & 15;
        async_b128(W + (size_t)e * D_MODEL + kb + c4 * 4,
                   Ws + e * LSTRIDE + c4 * 4);
    }
}

__global__ __launch_bounds__(BLOCK) void
moe_router_wmma(const float* __restrict__ x, const float* __restrict__ W,
                int* __restrict__ idx_out, float* __restrict__ w_out,
                float* __restrict__ probs_out, int n_tokens)
{
    // Two stage buffers for double-buffered async DMA; buffer 0 is reused
    // as the 64 x 65 logits scratch in the epilogue.
    __shared__ float smem[2 * BUF_FLOATS];   // 69,632 B

    const int tid  = threadIdx.x;
    const int lane = tid & 31;
    const int wid  = tid >> 5;            // 0..3, each wave owns a 16-token M-tile
    const int tok_base = blockIdx.x * M_TILE;

    const int mrow = lane & 15;           // M (A) / N (B) within the 16x16 tile
    const int half = lane >> 4;           // K={0,1} vs K={2,3} per f32 A/B VGPR layout

    v8f acc[4] = {};                      // 16 tokens x 64 experts

    // Prologue: stage 0 -> buffer 0
    issue_stage(x, W, tok_base, n_tokens, 0, smem, smem + M_TILE * LSTRIDE, tid);

    for (int s = 0; s < NSTAGES; ++s) {
        float* cur = smem + (s & 1) * BUF_FLOATS;
        float* nxt = smem + ((s + 1) & 1) * BUF_FLOATS;

        if (s + 1 < NSTAGES) {
            __syncthreads();   // everyone done reading `nxt` (holds stage s-1 data)
            issue_stage(x, W, tok_base, n_tokens, (s + 1) * KC,
                        nxt, nxt + M_TILE * LSTRIDE, tid);
            // Allow the 16 just-issued copies to stay in flight; everything
            // older (stage s's copies) must have landed.
            asm volatile("s_wait_asynccnt 0x10" ::: "memory");
        } else {
            asm volatile("s_wait_asynccnt 0x0" ::: "memory");
        }
        __syncthreads();       // stage s visible to all waves

        const float* xa = cur + (wid * 16 + mrow) * LSTRIDE + half * 2;
        const float* wb = cur + M_TILE * LSTRIDE + mrow * LSTRIDE + half * 2;
        #pragma unroll
        for (int kk = 0; kk < KC; kk += 4) {
            v2f a = *(const v2f*)(xa + kk);
            #pragma unroll
            for (int t = 0; t < 4; ++t) {
                v2f b = *(const v2f*)(wb + t * 16 * LSTRIDE + kk);
                acc[t] = __builtin_amdgcn_wmma_f32_16x16x4_f32(
                    false, a, false, b, (short)0, acc[t], false, false);
            }
        }
    }

    // ---- spill logits to LDS per the 16x16 f32 C/D layout ----
    // Last compute stage read buffer 1; buffer 0 was last read at stage
    // NSTAGES-2, and the barrier at s = NSTAGES-1 fenced those reads.
    float* Ls = smem; // 64 tokens x ESTRIDE
    #pragma unroll
    for (int t = 0; t < 4; ++t) {
        #pragma unroll
        for (int r = 0; r < 8; ++r) {
            int tl = wid * 16 + r + 8 * half;   // VGPR r: M=r (lanes 0-15), r+8 (16-31)
            int e  = t * 16 + mrow;
            Ls[tl * ESTRIDE + e] = acc[t][r];
        }
    }
    __syncthreads();

    // ---- softmax per token (one thread per token) ----
    if (tid < M_TILE) {
        float* row = Ls + tid * ESTRIDE;
        float mx = row[0];
        for (int j = 1; j < N_EXPERTS; ++j) mx = fmaxf(mx, row[j]);
        float sum = 0.0f;
        for (int j = 0; j < N_EXPERTS; ++j) {
            float e = expf(row[j] - mx);
            row[j] = e;
            sum += e;
        }
        float inv = 1.0f / sum;
        for (int j = 0; j < N_EXPERTS; ++j) row[j] *= inv;  // normalized probs
    }
    __syncthreads();

    // ---- coalesced probs write (before top-k mutates LDS) ----
    for (int i = tid; i < M_TILE * N_EXPERTS; i += BLOCK) {
        int r = i >> 6;
        int c = i & 63;
        int row = tok_base + r;
        if (row < n_tokens)
            probs_out[(size_t)row * N_EXPERTS + c] = Ls[r * ESTRIDE + c];
    }
    __syncthreads();

    // ---- top-8 + renormalized weights ----
    if (tid < M_TILE) {
        const int token = tok_base + tid;
        if (token < n_tokens) {
            float* row = Ls + tid * ESTRIDE;
            float vals[TOPK];
            int   ids[TOPK];
            float wsum = 0.0f;
            for (int k = 0; k < TOPK; ++k) {
                float best = -1.0f;   // probs >= 0
                int   bi   = 0;
                for (int j = 0; j < N_EXPERTS; ++j) {
                    float v = row[j];
                    if (v > best) { best = v; bi = j; }   // lowest index wins ties
                }
                vals[k] = best;
                ids[k]  = bi;
                wsum   += best;
                row[bi] = -1.0f;
            }
            float iw = 1.0f / (wsum + EPS);
            #pragma unroll
            for (int k = 0; k < TOPK; ++k) {
                idx_out[(size_t)token * TOPK + k] = ids[k];      // int32 bits
                w_out[(size_t)token * TOPK + k]   = vals[k] * iw;
            }
        }
    }
}

extern "C" void kernel_launch(void* const* d_in, const int* in_sizes, int n_in,
                              void* d_out, int out_size, void* d_ws, size_t ws_size,
                              hipStream_t stream) {
    const float* x = (const float*)d_in[0];  // [N, 4096] flattened (B*S collapsed)
    const float* W = (const float*)d_in[1];  // [64, 4096]
    const int n_tokens = in_sizes[0] / D_MODEL;   // 16384

    // d_out = [top_idx (N*8 as int32) | weights (N*8 f32) | probs (N*64 f32)]
    int*   idx_out   = (int*)d_out;
    float* w_out     = (float*)d_out + (size_t)n_tokens * TOPK;
    float* probs_out = (float*)d_out + (size_t)n_tokens * TOPK * 2;

    int grid = (n_tokens + M_TILE - 1) / M_TILE;   // 256 workgroups
    moe_router_wmma<<<grid, BLOCK, 0, stream>>>(x, W, idx_out, w_out, probs_out, n_tokens);
}